// HelixRecurrentBlock_13271448944888
// MI455X (gfx1250) — compile-verified
//
#include <hip/hip_runtime.h>
#include <math.h>

#define D_MODEL 1024
#define T_SEQ   4096
#define N_LOOPS 4
#define ACT_TH  0.99f
#define EPS_RMS 1e-6f

#define MBLK 128
#define NBLK 64
#define KC   16
#define LDSTR 18          // 16-DWORD tile row + 2 DWORDs TDM pad (8B-aligned, bank-spread)
#define NCHUNK (D_MODEL / KC)

typedef float v2f __attribute__((ext_vector_type(2)));
typedef float v8f __attribute__((ext_vector_type(8)));
typedef unsigned int u32x4 __attribute__((ext_vector_type(4)));
typedef int i32x4 __attribute__((ext_vector_type(4)));
typedef int i32x8 __attribute__((ext_vector_type(8)));

// ---------------------------------------------------------------------------
// TDM: issue a 2D tensor tile load (global -> LDS) with row padding.
// D# layout per CDNA5 ISA 8.3/8.4: group0 = {count, lds_addr, global_addr, type},
// group1 = {flags/pad, tensor dims, tile dims, stride0}.
// ---------------------------------------------------------------------------
__device__ __forceinline__ void tdm_load_2d(unsigned lds_addr, const float* gptr,
                                            unsigned tile_d0, unsigned tile_d1,
                                            unsigned tensor_d0, unsigned tensor_d1,
                                            unsigned stride0)
{
    unsigned long long ga = (unsigned long long)(uintptr_t)gptr;
    u32x4 g0;
    g0.x = 1u;                                              // count=1 valid descriptor
    g0.y = lds_addr;                                        // LDS byte address
    g0.z = (unsigned)ga;                                    // global_addr[31:0]
    g0.w = (unsigned)((ga >> 32) & 0x01FFFFFFu) | (2u << 30); // global_addr[56:32], type=2
    i32x8 g1;
    g1[0] = (int)((2u << 16)      // data_size = 4 bytes
                | (1u << 20)      // pad_enable (LDS dest padding)
                | (3u << 22)      // pad_interval: every 16 DWORDs
                | (1u << 25));    // pad_amount: 2 DWORDs
    g1[1] = (int)((tensor_d0 & 0xFFFFu) << 16);                       // tensor_dim0[15:0]
    g1[2] = (int)((tensor_d0 >> 16) | ((tensor_d1 & 0xFFFFu) << 16)); // dim0 hi / dim1 lo
    g1[3] = (int)((tensor_d1 >> 16) | (tile_d0 << 16));               // dim1 hi / tile_dim0
    g1[4] = (int)tile_d1;                                             // tile_dim1 (tile_dim2=0)
    g1[5] = (int)stride0;                                             // tensor_dim0_stride lo32
    g1[6] = 0;                                                        // stride0 hi16 / stride1 lo16
    g1[7] = 0;
    i32x4 z4 = {0, 0, 0, 0};
    i32x8 z8 = {0, 0, 0, 0, 0, 0, 0, 0};
    __builtin_amdgcn_tensor_load_to_lds(g0, g1, z4, z4, z8, 0);
}

// ---------------------------------------------------------------------------
// init: copy h into mutable workspace, zero h_out / cum_p / halted
// ---------------------------------------------------------------------------
__global__ void init_state(const float* __restrict__ h, float* __restrict__ h_ws,
                           float* __restrict__ out, float* __restrict__ cum_p,
                           float* __restrict__ halted, int total, int ntok) {
    int i = blockIdx.x * blockDim.x + threadIdx.x;
    if (i < total) { h_ws[i] = h[i]; out[i] = 0.f; }
    if (i < ntok)  { cum_p[i] = 0.f; halted[i] = 0.f; }
}

// ---------------------------------------------------------------------------
// one-time weight transpose: Wt[n][k] = W[k][n]  (makes B fragments contiguous)
// ---------------------------------------------------------------------------
__global__ __launch_bounds__(256) void transpose_w(const float* __restrict__ in,
                                                   float* __restrict__ out) {
    __shared__ float tile[32][33];
    const int bx = blockIdx.x * 32, by = blockIdx.y * 32;
    const int tx = threadIdx.x & 31, ty = threadIdx.x >> 5;   // 32 x 8
    #pragma unroll
    for (int r = 0; r < 32; r += 8)
        tile[ty + r][tx] = in[(size_t)(by + ty + r) * D_MODEL + bx + tx];
    __syncthreads();
    #pragma unroll
    for (int r = 0; r < 32; r += 8)
        out[(size_t)(bx + ty + r) * D_MODEL + by + tx] = tile[tx][ty + r];
}

// sinusoidal loop-index embedding, padded into D_MODEL
__device__ __forceinline__ float loop_buf(int t, int d) {
    if (d >= 64) return 0.f;
    int i = d & 31;
    float freq = expf(-(float)i * 0.03125f * 9.210340371976184f); // 10000^(-i/32)
    float ang  = (float)t * freq;
    return (d < 32) ? sinf(ang) : cosf(ang);
}

// ---------------------------------------------------------------------------
// K1: combined = rmsnorm(h + buf_t + e, rms_w)   one block per token
// ---------------------------------------------------------------------------
__global__ __launch_bounds__(256) void rmsnorm_combine(
    const float* __restrict__ h_ws, const float* __restrict__ e,
    const float* __restrict__ rms_w, float* __restrict__ combined, int t)
{
    __shared__ float red[256];
    const int tok = blockIdx.x;
    const int d   = threadIdx.x * 4;
    const size_t base = (size_t)tok * D_MODEL + d;

    float4 hv = *(const float4*)(h_ws + base);
    float4 ev = *(const float4*)(e + base);
    float4 x;
    x.x = hv.x + ev.x + loop_buf(t, d + 0);
    x.y = hv.y + ev.y + loop_buf(t, d + 1);
    x.z = hv.z + ev.z + loop_buf(t, d + 2);
    x.w = hv.w + ev.w + loop_buf(t, d + 3);

    red[threadIdx.x] = x.x * x.x + x.y * x.y + x.z * x.z + x.w * x.w;
    __syncthreads();
    for (int off = 128; off > 0; off >>= 1) {
        if (threadIdx.x < off) red[threadIdx.x] += red[threadIdx.x + off];
        __syncthreads();
    }
    const float rms = rsqrtf(red[0] * (1.0f / (float)D_MODEL) + EPS_RMS);

    float4 wv = *(const float4*)(rms_w + d);
    float4 o;
    o.x = x.x * rms * wv.x;
    o.y = x.y * rms * wv.y;
    o.z = x.z * rms * wv.z;
    o.w = x.w * rms * wv.w;
    *(float4*)(combined + base) = o;
}

// ---------------------------------------------------------------------------
// K2: dual-output FP32 WMMA GEMM with TDM double-buffered LDS staging.
//     Om = A*Wm, Ou = A*Wu ; Wmt/Wut are pre-transposed [n][k].
//     Block tile 128x64, 8 waves; wave w owns rows [16w,16w+16) x all 64 cols.
// ---------------------------------------------------------------------------
__global__ __launch_bounds__(256) void gemm_dual_wmma(
    const float* __restrict__ A, const float* __restrict__ Wmt,
    const float* __restrict__ Wut, float* __restrict__ Om,
    float* __restrict__ Ou, int nrows)
{
    __shared__ float As[2][MBLK * LDSTR];        // [buf][row][k(+pad)]
    __shared__ float Bs[2][2][NBLK * LDSTR];     // [buf][w][n][k(+pad)]

    const int tid   = threadIdx.x;
    const int wave  = tid >> 5;
    const int lane  = tid & 31;
    const int rowBase = blockIdx.x * MBLK;
    const int colBase = blockIdx.y * NBLK;
    const int waveRow = wave * 16;
    const int lrow    = lane & 15;
    const int khalf   = (lane >> 4) << 1;        // 0 or 2: K sub-pair per lane half
    const bool tdm_wave = (wave == 0);

    v8f acc[2][4];
    #pragma unroll
    for (int w = 0; w < 2; ++w)
        #pragma unroll
        for (int n = 0; n < 4; ++n) { v8f z = {}; acc[w][n] = z; }

    // TDM issue for one K-chunk into LDS buffer `buf`
    auto issue_chunk = [&](int k0, int buf) {
        tdm_load_2d((unsigned)(uintptr_t)&As[buf][0],
                    A + (size_t)rowBase * D_MODEL + k0,
                    KC, MBLK, D_MODEL, (unsigned)nrows, D_MODEL);
        tdm_load_2d((unsigned)(uintptr_t)&Bs[buf][0][0],
                    Wmt + (size_t)colBase * D_MODEL + k0,
                    KC, NBLK, D_MODEL, D_MODEL, D_MODEL);
        tdm_load_2d((unsigned)(uintptr_t)&Bs[buf][1][0],
                    Wut + (size_t)colBase * D_MODEL + k0,
                    KC, NBLK, D_MODEL, D_MODEL, D_MODEL);
    };

    if (tdm_wave) issue_chunk(0, 0);

    for (int i = 0; i < NCHUNK; ++i) {
        if (tdm_wave) __builtin_amdgcn_s_wait_tensorcnt(0);
        __syncthreads();                          // buffers i&1 now valid for all waves
        if (tdm_wave && (i + 1 < NCHUNK)) issue_chunk((i + 1) * KC, (i + 1) & 1);

        const float* Ab  = &As[i & 1][0];
        const float* Bb0 = &Bs[i & 1][0][0];
        const float* Bb1 = &Bs[i & 1][1][0];

        #pragma unroll
        for (int kk = 0; kk < KC; kk += 4) {
            // A fragment 16x4: lanes 0-15 K=kk+0/1, lanes 16-31 K=kk+2/3 (ISA 7.12.2)
            v2f a = *(const v2f*)(Ab + (waveRow + lrow) * LDSTR + kk + khalf);
            #pragma unroll
            for (int nt = 0; nt < 4; ++nt) {
                int n = nt * 16 + lrow;
                v2f bm = *(const v2f*)(Bb0 + n * LDSTR + kk + khalf);
                v2f bu = *(const v2f*)(Bb1 + n * LDSTR + kk + khalf);
                acc[0][nt] = __builtin_amdgcn_wmma_f32_16x16x4_f32(
                    false, a, false, bm, (short)0, acc[0][nt], false, false);
                acc[1][nt] = __builtin_amdgcn_wmma_f32_16x16x4_f32(
                    false, a, false, bu, (short)0, acc[1][nt], false, false);
            }
        }
        // no trailing barrier: next iteration's barrier guards buffer reuse
    }

    // C/D layout: VGPR r -> M = r (lanes 0-15) or r+8 (lanes 16-31), N = lane&15
    const int rowOff = rowBase + waveRow + ((lane >> 4) << 3);
    #pragma unroll
    for (int nt = 0; nt < 4; ++nt) {
        int col = colBase + nt * 16 + lrow;
        #pragma unroll
        for (int r = 0; r < 8; ++r) {
            Om[(size_t)(rowOff + r) * D_MODEL + col] = acc[0][nt][r];
            Ou[(size_t)(rowOff + r) * D_MODEL + col] = acc[1][nt][r];
        }
    }
}

// ---------------------------------------------------------------------------
// K3: helix aggregation + LTI update + halting head + ACT state
// ---------------------------------------------------------------------------
__global__ __launch_bounds__(256) void recurrent_update(
    const float* __restrict__ msg, const float* __restrict__ upd,
    const float* __restrict__ e, const float* __restrict__ log_A,
    const float* __restrict__ log_dt, const float* __restrict__ B_inj,
    const float* __restrict__ halt_w, const float* __restrict__ halt_b,
    float* __restrict__ h_ws, float* __restrict__ out,
    float* __restrict__ cum_p, float* __restrict__ halted)
{
    __shared__ float red[256];
    const int tok = blockIdx.x;
    const int ti  = tok & (T_SEQ - 1);
    const int d   = threadIdx.x * 4;
    const size_t base = (size_t)tok * D_MODEL + d;

    float4 agg = {0.f, 0.f, 0.f, 0.f};
    #pragma unroll
    for (int oi = 0; oi < 4; ++oi) {
        int o = 1 << oi;
        if (ti >= o) {
            float4 m = *(const float4*)(msg + base - (size_t)o * D_MODEL);
            agg.x += m.x; agg.y += m.y; agg.z += m.z; agg.w += m.w;
        }
    }

    float4 u  = *(const float4*)(upd + base);
    float4 hv = *(const float4*)(h_ws + base);
    float4 ev = *(const float4*)(e + base);
    float4 la = *(const float4*)(log_A + d);
    float4 bi = *(const float4*)(B_inj + d);
    float4 hw = *(const float4*)(halt_w + d);
    const float dt = log_dt[0];

    float4 Af;
    Af.x = expf(-expf(fminf(fmaxf(dt + la.x, -20.f), 20.f)));
    Af.y = expf(-expf(fminf(fmaxf(dt + la.y, -20.f), 20.f)));
    Af.z = expf(-expf(fminf(fmaxf(dt + la.z, -20.f), 20.f)));
    Af.w = expf(-expf(fminf(fmaxf(dt + la.w, -20.f), 20.f)));

    float4 hn;
    hn.x = Af.x * hv.x + bi.x * ev.x + u.x + 0.25f * agg.x;
    hn.y = Af.y * hv.y + bi.y * ev.y + u.y + 0.25f * agg.y;
    hn.z = Af.z * hv.z + bi.z * ev.z + u.z + 0.25f * agg.z;
    hn.w = Af.w * hv.w + bi.w * ev.w + u.w + 0.25f * agg.w;

    red[threadIdx.x] = hn.x * hw.x + hn.y * hw.y + hn.z * hw.z + hn.w * hw.w;
    __syncthreads();
    for (int off = 128; off > 0; off >>= 1) {
        if (threadIdx.x < off) red[threadIdx.x] += red[threadIdx.x + off];
        __syncthreads();
    }
    const float p = 1.f / (1.f + expf(-(red[0] + halt_b[0])));

    const float cp = cum_p[tok];
    const float hd = halted[tok];
    const float still = (hd != 0.f) ? 0.f : 1.f;
    const float rem = fmaxf(1.f - cp, 0.f);
    const float wgt = (((cp + p) >= ACT_TH) ? rem : p) * still;

    float4 ov = *(const float4*)(out + base);
    ov.x += wgt * hn.x; ov.y += wgt * hn.y; ov.z += wgt * hn.z; ov.w += wgt * hn.w;
    *(float4*)(out + base) = ov;
    *(float4*)(h_ws + base) = hn;

    if (threadIdx.x == 0) {
        float ncp = cp + p * still;
        cum_p[tok]  = ncp;
        halted[tok] = ((hd != 0.f) || (ncp >= ACT_TH)) ? 1.f : 0.f;
    }
}

// ---------------------------------------------------------------------------
extern "C" void kernel_launch(void* const* d_in, const int* in_sizes, int n_in,
                              void* d_out, int out_size, void* d_ws, size_t ws_size,
                              hipStream_t stream)
{
    const float* h      = (const float*)d_in[0];
    const float* e      = (const float*)d_in[1];
    const float* rms_w  = (const float*)d_in[2];
    const float* W_msg  = (const float*)d_in[3];
    const float* W_upd  = (const float*)d_in[4];
    const float* log_A  = (const float*)d_in[5];
    const float* log_dt = (const float*)d_in[6];
    const float* B_inj  = (const float*)d_in[7];
    const float* halt_w = (const float*)d_in[8];
    const float* halt_b = (const float*)d_in[9];
    float* out = (float*)d_out;

    const int total = in_sizes[0];              // B*T*D
    const int ntok  = total / D_MODEL;          // B*T

    float* ws       = (float*)d_ws;
    float* combined = ws;
    float* msg      = combined + (size_t)ntok * D_MODEL;
    float* upd      = msg      + (size_t)ntok * D_MODEL;
    float* h_ws     = upd      + (size_t)ntok * D_MODEL;
    float* Wmt      = h_ws     + (size_t)ntok * D_MODEL;
    float* Wut      = Wmt      + (size_t)D_MODEL * D_MODEL;
    float* cum_p    = Wut      + (size_t)D_MODEL * D_MODEL;
    float* halted   = cum_p    + ntok;

    init_state<<<dim3((total + 255) / 256), 256, 0, stream>>>(
        h, h_ws, out, cum_p, halted, total, ntok);

    dim3 tgrid(D_MODEL / 32, D_MODEL / 32);
    transpose_w<<<tgrid, 256, 0, stream>>>(W_msg, Wmt);
    transpose_w<<<tgrid, 256, 0, stream>>>(W_upd, Wut);

    dim3 ggrid(ntok / MBLK, D_MODEL / NBLK);
    for (int t = 0; t < N_LOOPS; ++t) {
        rmsnorm_combine<<<dim3(ntok), 256, 0, stream>>>(h_ws, e, rms_w, combined, t);
        gemm_dual_wmma<<<ggrid, 256, 0, stream>>>(combined, Wmt, Wut, msg, upd, ntok);
        recurrent_update<<<dim3(ntok), 256, 0, stream>>>(
            msg, upd, e, log_A, log_dt, B_inj, halt_w, halt_b,
            h_ws, out, cum_p, halted);
    }
}